// OntheshoulderGen_65841848648053
// MI455X (gfx1250) — compile-verified
//
#include <hip/hip_runtime.h>
#include <math.h>

#define NFRAMES_IN 62
#define NLM_IN     543
#define FEAT       252      // 84 * 3
#define NUM_FRAMES 15
#define OUT_PER    6804     // 6*504 + 15*252

typedef float v2f __attribute__((ext_vector_type(2)));
typedef float v8f __attribute__((ext_vector_type(8)));

// POINT_LANDMARKS = LIP(40) + [468..488] + [522..542]
__constant__ int c_pl[82] = {
  61,185,40,39,37,0,267,269,270,409,291,146,91,181,84,17,314,405,321,375,
  78,191,80,81,82,13,312,311,310,415,95,88,178,87,14,317,402,318,324,308,
  468,469,470,471,472,473,474,475,476,477,478,479,480,481,482,483,484,485,486,487,488,
  522,523,524,525,526,527,528,529,530,531,532,533,534,535,536,537,538,539,540,541,542
};

__device__ __forceinline__ float wave_sum(float v) {
#pragma unroll
  for (int off = 16; off >= 1; off >>= 1)
    v += __shfl_xor(v, off, 32);
  return v;
}

// Weight of original frame t in output column n.
// Padded frame list P[j] = clamp(j-1, 0, 61), j in [0,65); segment n covers j in [13n, 13n+13).
// Column 5 = overall mean (1/62). Columns >= 6 unused (zero).
__device__ __forceinline__ float seg_w(int t, int n) {
  if (t >= NFRAMES_IN || n >= 6) return 0.0f;
  if (n == 5) return 1.0f / 62.0f;
  int cnt = 0;
#pragma unroll
  for (int d = 0; d < 13; ++d) {
    int j = 13 * n + d - 1;
    j = j < 0 ? 0 : (j > 61 ? 61 : j);
    cnt += (j == t) ? 1 : 0;
  }
  return (float)cnt * (1.0f / 13.0f);
}

__global__ __launch_bounds__(256)
void feat_extract_kernel(const float* __restrict__ in, float* __restrict__ out) {
  __shared__ float xls[NFRAMES_IN * FEAT];   // 62 x 252 f32 = 62.5 KB

  const int tid  = threadIdx.x;
  const int lane = tid & 31;
  const int wave = tid >> 5;
  const float* src = in  + (long long)blockIdx.x * (NFRAMES_IN * NLM_IN * 3);
  float*       dst = out + (long long)blockIdx.x * OUT_PER;

  // ---- Phase A1: gather the 82 point landmarks into rows 2..83 --------------
  for (int i = tid; i < NFRAMES_IN * 82 * 3; i += 256) {
    int c = i % 3;
    int r = i / 3;
    int p = r % 82;
    int t = r / 82;
    int lm = c_pl[p];
    xls[t * FEAT + (p + 2) * 3 + c] = src[t * (NLM_IN * 3) + lm * 3 + c];
  }

  // ---- Phase A2: averaged rows 0 (lm 0..467) and 1 (lm 489..521) ------------
  for (int t = wave; t < NFRAMES_IN; t += 8) {
    if (t + 8 < NFRAMES_IN) {
      // stream-ahead hint -> global_prefetch_b8 (one cacheline per lane)
      __builtin_prefetch(src + (t + 8) * (NLM_IN * 3) + lane * 32, 0, 0);
    }
    float s0 = 0.f, s1 = 0.f, s2 = 0.f;
    for (int l = lane; l < 468; l += 32) {
      const float* p = src + t * (NLM_IN * 3) + l * 3;
      s0 += p[0]; s1 += p[1]; s2 += p[2];
    }
    s0 = wave_sum(s0); s1 = wave_sum(s1); s2 = wave_sum(s2);
    if (lane == 0) {
      xls[t * FEAT + 0] = s0 * (1.0f / 468.0f);
      xls[t * FEAT + 1] = s1 * (1.0f / 468.0f);
      xls[t * FEAT + 2] = s2 * (1.0f / 468.0f);
    }
    float q0 = 0.f, q1 = 0.f, q2 = 0.f;
    for (int l = 489 + lane; l < 522; l += 32) {
      const float* p = src + t * (NLM_IN * 3) + l * 3;
      q0 += p[0]; q1 += p[1]; q2 += p[2];
    }
    q0 = wave_sum(q0); q1 = wave_sum(q1); q2 = wave_sum(q2);
    if (lane == 0) {
      xls[t * FEAT + 3] = q0 * (1.0f / 33.0f);
      xls[t * FEAT + 4] = q1 * (1.0f / 33.0f);
      xls[t * FEAT + 5] = q2 * (1.0f / 33.0f);
    }
  }
  __syncthreads();

  // ---- Phase B: segment + overall mean/std via V_WMMA_F32_16X16X4_F32 -------
  // D(16x16) = A(16 feat x 4 time) x B(4 time x 16 cols).  Cols 0..4 = segment
  // means, col 5 = overall mean. Second accumulator chain on x^2 gives
  // std = sqrt(E[x^2] - m^2).
  const int nsel = lane & 15;        // B/D column index held by this lane
  const int krow = (lane >> 4) * 2;  // K (time) sub-row for this lane half
  v2f Bv[16];
#pragma unroll
  for (int kc = 0; kc < 16; ++kc) {
    int t0 = kc * 4 + krow;
    Bv[kc].x = seg_w(t0, nsel);
    Bv[kc].y = seg_w(t0 + 1, nsel);
  }

  const int mrow = lane & 15;        // A row (feature) held by this lane
  for (int ft = wave; ft < 16; ft += 8) {   // 16 feature tiles, 2 per wave
    int f  = ft * 16 + mrow;
    int fc = (f < FEAT) ? f : 0;     // tile 15 rows 252..255 are dead rows
    v8f acc1 = {};                   // sums of x * w
    v8f acc2 = {};                   // sums of x^2 * w
#pragma unroll
    for (int kc = 0; kc < 16; ++kc) {
      int t0 = kc * 4 + krow;
      v2f a;
      a.x = (t0     < NFRAMES_IN) ? xls[t0 * FEAT + fc]       : 0.0f;
      a.y = (t0 + 1 < NFRAMES_IN) ? xls[(t0 + 1) * FEAT + fc] : 0.0f;
      v2f a2; a2.x = a.x * a.x; a2.y = a.y * a.y;
      acc1 = __builtin_amdgcn_wmma_f32_16x16x4_f32(false, a,  false, Bv[kc],
                                                   (short)0, acc1, false, false);
      acc2 = __builtin_amdgcn_wmma_f32_16x16x4_f32(false, a2, false, Bv[kc],
                                                   (short)0, acc2, false, false);
    }
    // D layout: lane L, VGPR v  ->  M = v + 8*(L>=16), N = L&15
#pragma unroll
    for (int v = 0; v < 8; ++v) {
      int floc = v + 8 * (lane >> 4);
      int fo   = ft * 16 + floc;
      if (nsel < 6 && fo < FEAT) {
        float m   = acc1[v];
        float e2  = acc2[v];
        float var = e2 - m * m;
        var = var > 0.0f ? var : 0.0f;
        float sd = sqrtf(var);
        dst[nsel * 504 + fo]       = isfinite(m)  ? m  : 0.0f;
        dst[nsel * 504 + 252 + fo] = isfinite(sd) ? sd : 0.0f;
      }
    }
  }

  // ---- Phase C: bilinear temporal resize 62 -> 15 ---------------------------
  const float scale = 62.0f / 15.0f;
  for (int i = tid; i < NUM_FRAMES * FEAT; i += 256) {
    int k = i / FEAT;
    int f = i - k * FEAT;
    float s  = ((float)k + 0.5f) * scale - 0.5f;
    float fl = floorf(s);
    int i0 = (int)fl; i0 = i0 < 0 ? 0 : (i0 > 61 ? 61 : i0);
    int i1 = (i0 + 1 > 61) ? 61 : i0 + 1;
    float w = s - fl;
    float v = xls[i0 * FEAT + f] * (1.0f - w) + xls[i1 * FEAT + f] * w;
    dst[3024 + i] = isnan(v) ? 0.0f : v;
  }
}

extern "C" void kernel_launch(void* const* d_in, const int* in_sizes, int n_in,
                              void* d_out, int out_size, void* d_ws, size_t ws_size,
                              hipStream_t stream) {
  const float* x = (const float*)d_in[0];
  float* out = (float*)d_out;
  int batch = in_sizes[0] / (NFRAMES_IN * NLM_IN * 3);   // 2048
  feat_extract_kernel<<<dim3(batch), dim3(256), 0, stream>>>(x, out);
}